// TorchConstantVelocityODKalmanFilter_48155173323364
// MI455X (gfx1250) — compile-verified
//
#include <hip/hip_runtime.h>

typedef __attribute__((ext_vector_type(2))) float v2f;
typedef __attribute__((ext_vector_type(8))) float v8f;

__device__ __forceinline__ float lane_bcast(float v, int srcLane) {
    return __shfl(v, srcLane, 32);
}

// One wave32 runs the whole sequential Kalman scan.
// P (8x8, symmetric) is embedded in the WMMA 16x16 f32 C/D register layout:
//   c[m] at lane n holds P[m + 8*(n>=16), n & 15]; rows/cols >= 8 stay zero.
__global__ __launch_bounds__(32)
void kalman_scan_kernel(const float* __restrict__ z,      // (N,1,4)
                        const float* __restrict__ t_obs,  // (N,1,1)
                        const float* __restrict__ t_un,   // (M,1,1)
                        float* __restrict__ out,          // (M,1,4)
                        int N, int M) {
    const int  lane    = (int)threadIdx.x;   // 0..31
    const bool lowHalf = lane < 16;
    const int  col     = lane & 15;

    // ---- initial state: xs = 0, P = P0 * I8 ----
    v8f c;
#pragma unroll
    for (int m = 0; m < 8; ++m) c[m] = (lane == m) ? 10.0f : 0.0f;

    float xs[8];
#pragma unroll
    for (int r = 0; r < 8; ++r) xs[r] = 0.0f;

    const float colSel = (col < 4) ? 1.0f : 0.0f;   // columns that receive +dt*col(+4)

    for (int i = 0; i < N; ++i) {
        const float dt = (i + 1 < N) ? (t_obs[i + 1] - t_obs[i]) : 1.0f;

        // ================= predict: xs = F xs ; P = F P F^T + Q =================
#pragma unroll
        for (int r = 0; r < 4; ++r) xs[r] = fmaf(dt, xs[r + 4], xs[r]);

        // row shift: rows 0..3 += dt * rows 4..7 (in-register)
#pragma unroll
        for (int m = 0; m < 4; ++m) c[m] = fmaf(dt, c[m + 4], c[m]);

        // col shift: cols 0..3 += dt * cols 4..7 (cross-lane)
        const float cmask = colSel * dt;
#pragma unroll
        for (int m = 0; m < 8; ++m) {
            float t = lane_bcast(c[m], (lane + 4) & 31);
            c[m] = fmaf(cmask, t, c[m]);
        }
        // + Q_STD^2 * I on the 8x8 diagonal
#pragma unroll
        for (int m = 0; m < 8; ++m) c[m] += (lane == m) ? 1.0f : 0.0f;

        // ================= update =================
        // S = P[:4,:4] + R^2 I  (uniform scalars in every lane)
        float S[4][4];
#pragma unroll
        for (int k = 0; k < 4; ++k)
#pragma unroll
            for (int j = 0; j < 4; ++j)
                S[k][j] = lane_bcast(c[k], j) + ((k == j) ? 1.0f : 0.0f);

        // S^-1 via fully unrolled Gauss-Jordan (SPD: diagonal pivots are safe)
        float A[4][8];
#pragma unroll
        for (int r = 0; r < 4; ++r) {
#pragma unroll
            for (int cc = 0; cc < 4; ++cc) A[r][cc] = S[r][cc];
#pragma unroll
            for (int cc = 0; cc < 4; ++cc) A[r][4 + cc] = (r == cc) ? 1.0f : 0.0f;
        }
#pragma unroll
        for (int k = 0; k < 4; ++k) {
            float p = 1.0f / A[k][k];
#pragma unroll
            for (int cc = 0; cc < 8; ++cc) A[k][cc] *= p;
#pragma unroll
            for (int r = 0; r < 4; ++r) {
                if (r == k) continue;
                float f = A[r][k];
#pragma unroll
                for (int cc = 0; cc < 8; ++cc) A[r][cc] = fmaf(-f, A[k][cc], A[r][cc]);
            }
        }
        float si[4][4];
#pragma unroll
        for (int k = 0; k < 4; ++k)
#pragma unroll
            for (int j = 0; j < 4; ++j) si[k][j] = A[k][4 + j];

        // innovation y = z - xs[:4]; w = S^-1 y   (uniform)
        const float* zrow = z + 4 * i;
        float y[4];
#pragma unroll
        for (int j = 0; j < 4; ++j) y[j] = zrow[j] - xs[j];
        float w[4];
#pragma unroll
        for (int k = 0; k < 4; ++k) {
            float acc = 0.0f;
#pragma unroll
            for (int j = 0; j < 4; ++j) acc = fmaf(si[k][j], y[j], acc);
            w[k] = acc;
        }

        // xs += K y = P[:,:4] w : per-lane t(n) = sum_k P[k,n]*w[k] = dxs[n] (symmetry)
        float tacc = 0.0f;
#pragma unroll
        for (int k = 0; k < 4; ++k) tacc = fmaf(c[k], w[k], tacc);
#pragma unroll
        for (int r = 0; r < 8; ++r) xs[r] += lane_bcast(tacc, r);

        // cP[j] = P[j, lane&15] in all 32 lanes (broadcast low half to high half)
        float cP[4];
#pragma unroll
        for (int j = 0; j < 4; ++j) cP[j] = lane_bcast(c[j], col);

        // A operand (16x4, A-layout) = P[:, 0:4] using symmetry P[r,k] = P[k,r]:
        //   lanes 0..15 : row r = lane,    cols {0,1}
        //   lanes 16..31: row r = lane-16, cols {2,3}
        v2f a;
        a[0] = lowHalf ? cP[0] : cP[2];
        a[1] = lowHalf ? cP[1] : cP[3];

        // B operand (4x16, B-layout) = -(S^-1 @ P[:4,:]):
        //   VGPR0: rows {0 | 2}, VGPR1: rows {1 | 3} across lane halves
        float b0 = 0.0f, b1 = 0.0f;
#pragma unroll
        for (int j = 0; j < 4; ++j) {
            float s0 = lowHalf ? si[0][j] : si[2][j];
            float s1 = lowHalf ? si[1][j] : si[3][j];
            b0 = fmaf(s0, cP[j], b0);
            b1 = fmaf(s1, cP[j], b1);
        }
        v2f b;
        b[0] = -b0;
        b[1] = -b1;

        // P <- P - P[:,0:4] @ (S^-1 @ P[0:4,:])  ==  P - K @ P[:4,:]
        c = __builtin_amdgcn_wmma_f32_16x16x4_f32(
                /*neg_a=*/false, a, /*neg_b=*/false, b,
                /*c_mod=*/(short)0, c, /*reuse_a=*/false, /*reuse_b=*/false);
    }

    // ================= unobserved predictions (xs only; P unused for output) ====
    const float tlast = t_obs[N - 1];
    for (int i = 0; i < M; ++i) {
        const float dt = t_un[i] - tlast;
#pragma unroll
        for (int r = 0; r < 4; ++r) xs[r] = fmaf(dt, xs[r + 4], xs[r]);
        if (lane == 0) {
            out[4 * i + 0] = xs[0];
            out[4 * i + 1] = xs[1];
            out[4 * i + 2] = xs[2];
            out[4 * i + 3] = xs[3];
        }
    }
}

extern "C" void kernel_launch(void* const* d_in, const int* in_sizes, int n_in,
                              void* d_out, int out_size, void* d_ws, size_t ws_size,
                              hipStream_t stream) {
    (void)n_in; (void)d_ws; (void)ws_size; (void)out_size;
    const float* x      = (const float*)d_in[0];  // (N,1,4)
    const float* t_obs  = (const float*)d_in[1];  // (N,1,1)
    const float* t_un   = (const float*)d_in[2];  // (M,1,1)
    float*       out    = (float*)d_out;          // (M,1,4)
    const int N = in_sizes[1];
    const int M = in_sizes[2];
    kalman_scan_kernel<<<1, 32, 0, stream>>>(x, t_obs, t_un, out, N, M);
}